// RAMultiheadAttention_22720376995948
// MI455X (gfx1250) — compile-verified
//
#include <hip/hip_runtime.h>
#include <hip/hip_bf16.h>

// MI455X / gfx1250, wave32. All matmuls use V_WMMA_F32_16X16X4_F32 (fp32,
// full precision) because this workload is HBM-bound (~0.8 GB traffic,
// ~26 GFLOP => ~0.75 TFLOP/s needed at 23.3 TB/s): the memory system, not
// the matrix pipe, is the limiter, so fp32 WMMA is the right choice.
//
// The two 256 MB relation tensors are streamed exactly once each, with
// non-temporal load hints (TH=NT) so they do not evict the reused working
// set (qkv 25 MB, scores 64 MB, weights) from the 192 MB global L2.

typedef float v2f __attribute__((ext_vector_type(2)));
typedef float v8f __attribute__((ext_vector_type(8)));

#define T_LEN 512
#define BSZ   4
#define EMB   1024
#define HEADS 16
#define HDIM  64
#define M_ROWS (T_LEN * BSZ)   // 2048 (t*B+b flattened rows)

__device__ __forceinline__ v8f wmma_f32_k4(v2f a, v2f b, v8f c) {
    // (neg_a, A, neg_b, B, c_mod, C, reuse_a, reuse_b); f32 has no A/B neg.
    return __builtin_amdgcn_wmma_f32_16x16x4_f32(
        false, a, false, b, (short)0, c, false, false);
}

// ---------------------------------------------------------------------------
// K1: qkv[m][f] = x[m][:] . in_proj_w[f][:] + bias[f];  q part scaled by 1/8.
//     m = t*B + b (x is [T,B,E] so rows are already contiguous).
//     Tiles: 128 (M) x 192 (N) = 24576, one per wave.
// ---------------------------------------------------------------------------
__global__ __launch_bounds__(256) void k_qkv(const float* __restrict__ x,
                                             const float* __restrict__ w,
                                             const float* __restrict__ bias,
                                             float* __restrict__ qkv) {
    int tile = blockIdx.x * 8 + (threadIdx.x >> 5);
    const int NT = (3 * EMB) / 16;              // 192
    int mt = (tile / NT) * 16;
    int nt = (tile % NT) * 16;
    int lane = threadIdx.x & 31;
    int row = lane & 15, koff = (lane >> 4) * 2;

    const float* arow = x + (size_t)(mt + row) * EMB;
    const float* brow = w + (size_t)(nt + row) * EMB;
    v8f acc = {};
    for (int kk = 0; kk < EMB; kk += 4) {
        v2f a = *(const v2f*)(arow + kk + koff);
        v2f b = *(const v2f*)(brow + kk + koff);
        acc = wmma_f32_k4(a, b, acc);
    }
    int n = lane & 15, mbase = (lane >> 4) * 8;
    for (int r = 0; r < 8; ++r) {
        int m = mt + mbase + r;
        int f = nt + n;
        float v = acc[r] + bias[f];
        if (f < EMB) v *= 0.125f;               // d^-0.5, applied to q only
        qkv[(size_t)m * (3 * EMB) + f] = v;
    }
}

// ---------------------------------------------------------------------------
// K2: content scores S[b][h][q][k] = q . k   (per b,h GEMM 512x512x64)
//     Tiles: b(2) h(4) qt(5) kt(5) -> 65536 waves.
// ---------------------------------------------------------------------------
__global__ __launch_bounds__(256) void k_scores(const float* __restrict__ qkv,
                                                float* __restrict__ S) {
    int tile = blockIdx.x * 8 + (threadIdx.x >> 5);
    int kt = (tile & 31) * 16; tile >>= 5;
    int qt = (tile & 31) * 16; tile >>= 5;
    int h  = tile & 15;        tile >>= 4;
    int b  = tile;
    int lane = threadIdx.x & 31;
    int row = lane & 15, koff = (lane >> 4) * 2;

    const float* qbase = qkv + ((size_t)(qt + row) * BSZ + b) * (3 * EMB) + h * HDIM;
    const float* kbase = qkv + ((size_t)(kt + row) * BSZ + b) * (3 * EMB) + EMB + h * HDIM;
    v8f acc = {};
    for (int kk = 0; kk < HDIM; kk += 4) {
        v2f a  = *(const v2f*)(qbase + kk + koff);
        v2f bb = *(const v2f*)(kbase + kk + koff);
        acc = wmma_f32_k4(a, bb, acc);
    }
    float* out = S + (((size_t)b * HEADS + h) * T_LEN + qt) * T_LEN + kt;
    int n = lane & 15, mbase = (lane >> 4) * 8;
    for (int r = 0; r < 8; ++r)
        out[(size_t)(mbase + r) * T_LEN + n] = acc[r];
}

// ---------------------------------------------------------------------------
// K3: relation-K bias. For fixed (b,q): Bias[k,h] = rel_k[b,q] (Tx64) . Q^T.
//     Streams the 256 MB rel_k exactly once (non-temporal).
//     Tiles: b(2) q(9) kt(5) -> 65536.  S[b][h][q][k] += Bias (after K2).
// ---------------------------------------------------------------------------
__global__ __launch_bounds__(256) void k_bias(const float* __restrict__ qkv,
                                              const float* __restrict__ relk,
                                              float* __restrict__ S) {
    int tile = blockIdx.x * 8 + (threadIdx.x >> 5);
    int kt = (tile & 31) * 16; tile >>= 5;
    int q  = tile & 511;       tile >>= 9;
    int b  = tile;
    int lane = threadIdx.x & 31;
    int row = lane & 15, koff = (lane >> 4) * 2;

    const float* arow = relk + (((size_t)b * T_LEN + q) * T_LEN + kt + row) * HDIM;
    // B[c][n] = Qscaled[b, head=n, q, c] -> fetched transposed, contiguous in c
    const float* brow = qkv + ((size_t)q * BSZ + b) * (3 * EMB) + (size_t)row * HDIM;
    v8f acc = {};
    for (int kk = 0; kk < HDIM; kk += 4) {
        v2f a  = __builtin_nontemporal_load((const v2f*)(arow + kk + koff));
        v2f bb = *(const v2f*)(brow + kk + koff);
        acc = wmma_f32_k4(a, bb, acc);
    }
    int n = lane & 15, mbase = (lane >> 4) * 8;
    for (int r = 0; r < 8; ++r) {
        int k = kt + mbase + r;   // tile M dim = key index
        int h = n;                // tile N dim = head
        size_t idx = (((size_t)b * HEADS + h) * T_LEN + q) * T_LEN + k;
        S[idx] += acc[r];
    }
}

// ---------------------------------------------------------------------------
// K4: softmax over k, one wave per (b,h,q) row of 512; 16 elems/lane.
// ---------------------------------------------------------------------------
__global__ __launch_bounds__(256) void k_softmax(float* __restrict__ S) {
    int row = blockIdx.x * 8 + (threadIdx.x >> 5);
    int lane = threadIdx.x & 31;
    float* p = S + (size_t)row * T_LEN;
    float vals[16];
    float m = -__builtin_inff();
    for (int i = 0; i < 16; ++i) { vals[i] = p[lane + i * 32]; m = fmaxf(m, vals[i]); }
    for (int o = 16; o; o >>= 1) m = fmaxf(m, __shfl_xor(m, o, 32));
    float s = 0.0f;
    for (int i = 0; i < 16; ++i) { vals[i] = __expf(vals[i] - m); s += vals[i]; }
    for (int o = 16; o; o >>= 1) s += __shfl_xor(s, o, 32);
    float inv = 1.0f / s;
    for (int i = 0; i < 16; ++i) p[lane + i * 32] = vals[i] * inv;
}

// ---------------------------------------------------------------------------
// K5: ctx = attn . V per (b,h); written directly in [t,b,e] layout.
//     Tiles: b(2) h(4) qt(5) dt(2) -> 8192.
// ---------------------------------------------------------------------------
__global__ __launch_bounds__(256) void k_ctx(const float* __restrict__ S,
                                             const float* __restrict__ qkv,
                                             float* __restrict__ ctx) {
    int tile = blockIdx.x * 8 + (threadIdx.x >> 5);
    int dt = (tile & 3) * 16;  tile >>= 2;
    int qt = (tile & 31) * 16; tile >>= 5;
    int h  = tile & 15;        tile >>= 4;
    int b  = tile;
    int lane = threadIdx.x & 31;
    int row = lane & 15, koff = (lane >> 4) * 2;

    const float* arow = S + (((size_t)b * HEADS + h) * T_LEN + qt + row) * T_LEN;
    // B[c][n] = V[b,h,c,dt+n] = qkv[(c*B+b)*3E + 2E + h*64 + dt + n]
    const float* bcol = qkv + 2 * EMB + (size_t)h * HDIM + dt + (lane & 15)
                            + (size_t)b * (3 * EMB);
    const size_t rstride = (size_t)BSZ * 3 * EMB;
    v8f acc = {};
    for (int kk = 0; kk < T_LEN; kk += 4) {
        v2f a = *(const v2f*)(arow + kk + koff);
        v2f bb;
        bb[0] = bcol[(size_t)(kk + koff)     * rstride];
        bb[1] = bcol[(size_t)(kk + koff + 1) * rstride];
        acc = wmma_f32_k4(a, bb, acc);
    }
    int n = lane & 15, mbase = (lane >> 4) * 8;
    for (int r = 0; r < 8; ++r) {
        int q = qt + mbase + r;
        ctx[((size_t)q * BSZ + b) * EMB + h * HDIM + dt + n] = acc[r];
    }
}

// ---------------------------------------------------------------------------
// K6: ctx += attn . rel_v. For fixed (b,q): O[h,d] = A[h,:] (HxT) . relv[b,q].
//     Streams the 256 MB rel_v exactly once (non-temporal).
//     Tiles: b(2) q(9) dt(2) -> 8192.
// ---------------------------------------------------------------------------
__global__ __launch_bounds__(256) void k_ctx_rel(const float* __restrict__ S,
                                                 const float* __restrict__ relv,
                                                 float* __restrict__ ctx) {
    int tile = blockIdx.x * 8 + (threadIdx.x >> 5);
    int dt = (tile & 3) * 16; tile >>= 2;
    int q  = tile & 511;      tile >>= 9;
    int b  = tile;
    int lane = threadIdx.x & 31;
    int row = lane & 15, koff = (lane >> 4) * 2;

    // A[m=h][c] = attn[b, h, q, c] (contiguous in c)
    const float* arow = S + (((size_t)b * HEADS + row) * T_LEN + q) * T_LEN;
    const float* bcol = relv + (((size_t)b * T_LEN + q) * T_LEN) * HDIM + dt + (lane & 15);
    v8f acc = {};
    for (int kk = 0; kk < T_LEN; kk += 4) {
        v2f a = *(const v2f*)(arow + kk + koff);
        v2f bb;
        bb[0] = __builtin_nontemporal_load(bcol + (size_t)(kk + koff)     * HDIM);
        bb[1] = __builtin_nontemporal_load(bcol + (size_t)(kk + koff + 1) * HDIM);
        acc = wmma_f32_k4(a, bb, acc);
    }
    int n = lane & 15, mbase = (lane >> 4) * 8;
    for (int r = 0; r < 8; ++r) {
        int h = mbase + r;
        size_t idx = ((size_t)q * BSZ + b) * EMB + h * HDIM + dt + n;
        ctx[idx] += acc[r];
    }
}

// ---------------------------------------------------------------------------
// K7: out[m][f] = ctx[m][:] . out_w[f][:] + out_b[f].  Tiles mt(7) nt(6).
// ---------------------------------------------------------------------------
__global__ __launch_bounds__(256) void k_outproj(const float* __restrict__ ctx,
                                                 const float* __restrict__ w,
                                                 const float* __restrict__ bias,
                                                 float* __restrict__ out) {
    int tile = blockIdx.x * 8 + (threadIdx.x >> 5);
    int nt = (tile & 63) * 16; tile >>= 6;
    int mt = tile * 16;
    int lane = threadIdx.x & 31;
    int row = lane & 15, koff = (lane >> 4) * 2;

    const float* arow = ctx + (size_t)(mt + row) * EMB;
    const float* brow = w   + (size_t)(nt + row) * EMB;
    v8f acc = {};
    for (int kk = 0; kk < EMB; kk += 4) {
        v2f a  = *(const v2f*)(arow + kk + koff);
        v2f bb = *(const v2f*)(brow + kk + koff);
        acc = wmma_f32_k4(a, bb, acc);
    }
    int n = lane & 15, mbase = (lane >> 4) * 8;
    for (int r = 0; r < 8; ++r)
        out[(size_t)(mt + mbase + r) * EMB + nt + n] = acc[r] + bias[nt + n];
}

// ---------------------------------------------------------------------------
extern "C" void kernel_launch(void* const* d_in, const int* in_sizes, int n_in,
                              void* d_out, int out_size, void* d_ws, size_t ws_size,
                              hipStream_t stream) {
    const float* x     = (const float*)d_in[0];   // [T,B,E]
    const float* relk  = (const float*)d_in[1];   // [B,T,T,d]
    const float* relv  = (const float*)d_in[2];   // [B,T,T,d]
    const float* win   = (const float*)d_in[3];   // [3E,E]
    const float* bin   = (const float*)d_in[4];   // [3E]
    const float* wout  = (const float*)d_in[5];   // [E,E]
    const float* bout  = (const float*)d_in[6];   // [E]
    float* out = (float*)d_out;

    float* ws  = (float*)d_ws;
    float* qkv = ws;                                              // 2048*3072 f
    float* S   = qkv + (size_t)M_ROWS * 3 * EMB;                  // 4*16*512*512 f
    float* ctx = S + (size_t)BSZ * HEADS * T_LEN * T_LEN;         // 2048*1024 f

    dim3 blk(256);
    k_qkv    <<<3072, blk, 0, stream>>>(x, win, bin, qkv);
    k_scores <<<8192, blk, 0, stream>>>(qkv, S);
    k_bias   <<<8192, blk, 0, stream>>>(qkv, relk, S);
    k_softmax<<<4096, blk, 0, stream>>>(S);
    k_ctx    <<<1024, blk, 0, stream>>>(S, qkv, ctx);
    k_ctx_rel<<<1024, blk, 0, stream>>>(S, relv, ctx);
    k_outproj<<<1024, blk, 0, stream>>>(ctx, wout, bout, out);
}